// HybridSegmentationLoss_50208167690243
// MI455X (gfx1250) — compile-verified
//
#include <hip/hip_runtime.h>
#include <hip/hip_bf16.h>
#include <math.h>

// Problem constants (B,C,H,W) = (16,1,512,512)
#define NB_   16
#define NH_   512
#define NW_   512
#define HW_   (NH_ * NW_)          // 262144
#define NTOT_ (NB_ * HW_)          // 4194304
#define BIG_  2048.0f              // W*4 "infinity" from the reference scan init

typedef __attribute__((ext_vector_type(2))) float v2f;
typedef __attribute__((ext_vector_type(8))) float v8f;

// ---------------------------------------------------------------------------
// Wave-wide f32 sum via V_WMMA_F32_16X16X4_F32 (codegen-verified round 1).
// A (16x4): VGPR0 = lanes0-15 -> A[m,0], lanes16-31 -> A[m,2]; VGPR1 zeroed.
// B = all ones => D[m,n] = x[m] + x[m+16], independent of n and of B striping.
// Requires EXEC all-ones: call only from fully-uniform code paths.
// ---------------------------------------------------------------------------
__device__ __forceinline__ float wave_sum32(float x) {
  v2f a; a[0] = x;    a[1] = 0.0f;
  v2f b; b[0] = 1.0f; b[1] = 1.0f;
  v8f c = {};
  v8f d = __builtin_amdgcn_wmma_f32_16x16x4_f32(
      /*neg_a=*/false, a, /*neg_b=*/false, b,
      /*c_mod=*/(short)0, c, /*reuse_a=*/false, /*reuse_b=*/false);
  float s = d[0] + d[1] + d[2] + d[3] + d[4] + d[5] + d[6] + d[7];
  return __shfl(s, 0) + __shfl(s, 16);
}

// ---------------------------------------------------------------------------
// K1: 1D run-length distances along W (forward + backward), pos & neg masks.
// One wave per row; ballot + clz/ffs gives each 32-chunk in O(1) per lane.
// Writes squared min(f,b) into g2pos / g2neg.
// ---------------------------------------------------------------------------
__global__ void k_rowdist(const int* __restrict__ tgt,
                          float* __restrict__ g2pos,
                          float* __restrict__ g2neg) {
  const int lane = threadIdx.x & 31;
  const int wave = threadIdx.x >> 5;
  const int row  = blockIdx.x * 8 + wave;       // 8192 rows total
  const size_t base = (size_t)row * NW_;

  // Forward scan: d = fg ? d+1 : 0, seeded with BIG_.
  float cp = BIG_, cn = BIG_;
  for (int ch = 0; ch < 16; ++ch) {
    const int i = ch * 32 + lane;
    const int t = tgt[base + i];
    const bool fp = (t != 0);
    const unsigned mrp = (unsigned)__ballot(!fp);   // resets for pos scan
    const unsigned mrn = (unsigned)__ballot(fp);    // resets for neg scan
    float dp, dn;
    {
      const unsigned pre = mrp & (0xFFFFFFFFu >> (31 - lane));
      if (!fp)      dp = 0.0f;
      else if (pre) dp = (float)(lane - (31 - __clz((int)pre)));
      else          dp = cp + (float)(lane + 1);
    }
    {
      const unsigned pre = mrn & (0xFFFFFFFFu >> (31 - lane));
      if (fp)       dn = 0.0f;
      else if (pre) dn = (float)(lane - (31 - __clz((int)pre)));
      else          dn = cn + (float)(lane + 1);
    }
    cp = __shfl(dp, 31);
    cn = __shfl(dn, 31);
    g2pos[base + i] = dp;
    g2neg[base + i] = dn;
  }

  // Backward scan, then g2 = min(f,b)^2.
  cp = BIG_; cn = BIG_;
  for (int ch = 15; ch >= 0; --ch) {
    const int i = ch * 32 + lane;
    const int t = tgt[base + i];
    const bool fp = (t != 0);
    const unsigned mrp = (unsigned)__ballot(!fp);
    const unsigned mrn = (unsigned)__ballot(fp);
    float bp, bn;
    {
      const unsigned post = mrp & (0xFFFFFFFFu << lane);
      if (!fp)       bp = 0.0f;
      else if (post) bp = (float)((__ffs((int)post) - 1) - lane);
      else           bp = cp + (float)(32 - lane);
    }
    {
      const unsigned post = mrn & (0xFFFFFFFFu << lane);
      if (fp)        bn = 0.0f;
      else if (post) bn = (float)((__ffs((int)post) - 1) - lane);
      else           bn = cn + (float)(32 - lane);
    }
    cp = __shfl(bp, 0);
    cn = __shfl(bn, 0);
    const float gp = fminf(g2pos[base + i], bp);
    const float gn = fminf(g2neg[base + i], bn);
    g2pos[base + i] = gp * gp;
    g2neg[base + i] = gn * gn;
  }
}

// ---------------------------------------------------------------------------
// K2: column lower envelope d2[x,w] = min_z (g2[z,w] + (z-x)^2).
// Block = 256 threads -> 32 columns x 32 x-rows tile (4 x per thread).
// g2 chunks (64 z x 32 w = 8 KB) DMA'd straight into LDS with
// GLOBAL_LOAD_ASYNC_TO_LDS_B128 (ASYNCcnt path), then fma/min inner loop.
// grid.z = mask*16 + img (0..31).
// ---------------------------------------------------------------------------
__global__ void k_envelope(const float* __restrict__ g2,
                           float* __restrict__ d2) {
  const int tid = threadIdx.x;
  const int w0  = blockIdx.x * 32;
  const int x0  = blockIdx.y * 32;
  const size_t base = (size_t)blockIdx.z * HW_;
  const int tw = tid & 31;      // column within tile
  const int tg = tid >> 5;      // x-group 0..7 (4 x each)

  __shared__ __attribute__((aligned(16))) float sg[64 * 32];

  float dmin0 = 3.0e38f, dmin1 = 3.0e38f, dmin2 = 3.0e38f, dmin3 = 3.0e38f;
  const float xb = (float)(x0 + tg * 4);
  float dz0 = 0.0f - xb;
  float dz1 = dz0 - 1.0f;
  float dz2 = dz0 - 2.0f;
  float dz3 = dz0 - 3.0f;

  for (int zc = 0; zc < NH_; zc += 64) {
    __syncthreads();   // previous chunk fully consumed before refill
    // Async DMA: 2 x (256 lanes x 16B) = 8 KB chunk, global -> LDS.
    // Group g covers 4 consecutive floats: zl = g>>3, w-offset = (g&7)*4.
#pragma unroll
    for (int r = 0; r < 2; ++r) {
      const int g  = tid + r * 256;           // 0..511
      const int zl = g >> 3;
      const int wl = (g & 7) * 4;
      const unsigned ldsoff =
          (unsigned)(uintptr_t)(const void*)&sg[zl * 32 + wl];
      const void* gptr =
          (const void*)&g2[base + (size_t)(zc + zl) * NW_ + (w0 + wl)];
      asm volatile("global_load_async_to_lds_b128 %0, %1, off"
                   :
                   : "v"(ldsoff), "v"(gptr)
                   : "memory");
    }
    if (zc + 64 < NH_) {
      // gfx1250 global_prefetch_b8 for the next z-chunk (L2 warm-up)
      __builtin_prefetch(&g2[base + (size_t)(zc + 64 + tg) * NW_ + (w0 + tw)], 0, 1);
    }
    // Own wave's async transfers complete, then block-wide visibility.
    asm volatile("s_wait_asynccnt 0x0" ::: "memory");
    __syncthreads();
#pragma unroll 4
    for (int zz = 0; zz < 64; ++zz) {
      const float q = sg[zz * 32 + tw];
      dmin0 = fminf(dmin0, fmaf(dz0, dz0, q)); dz0 += 1.0f;
      dmin1 = fminf(dmin1, fmaf(dz1, dz1, q)); dz1 += 1.0f;
      dmin2 = fminf(dmin2, fmaf(dz2, dz2, q)); dz2 += 1.0f;
      dmin3 = fminf(dmin3, fmaf(dz3, dz3, q)); dz3 += 1.0f;
    }
  }

  const int x = x0 + tg * 4;
  d2[base + (size_t)(x + 0) * NW_ + w0 + tw] = dmin0;
  d2[base + (size_t)(x + 1) * NW_ + w0 + tw] = dmin1;
  d2[base + (size_t)(x + 2) * NW_ + w0 + tw] = dmin2;
  d2[base + (size_t)(x + 3) * NW_ + w0 + tw] = dmin3;
}

// ---------------------------------------------------------------------------
// K3: per-image bmax = max(sqrt(pos2)+sqrt(neg2)) and anypos;
// writes scale[img] = anypos ? 1/(bmax+1e-8) : 0.
// ---------------------------------------------------------------------------
__global__ void k_imgstats(const int* __restrict__ tgt,
                           const float* __restrict__ d2,
                           float* __restrict__ scale) {
  const int img = blockIdx.x;
  const int tid = threadIdx.x;
  const float* p2 = d2 + (size_t)img * HW_;
  const float* n2 = d2 + (size_t)(NB_ + img) * HW_;
  const int*   tg = tgt + (size_t)img * HW_;

  float vmax = 0.0f, tsum = 0.0f;
  for (int i = tid; i < HW_; i += 256) {
    const float bnd = sqrtf(p2[i]) + sqrtf(n2[i]);
    vmax = fmaxf(vmax, bnd);
    tsum += (tg[i] != 0) ? 1.0f : 0.0f;
  }
  tsum = wave_sum32(tsum);   // uniform path: EXEC all ones

  __shared__ float smax[256];
  __shared__ float ssum[8];
  smax[tid] = vmax;
  if ((tid & 31) == 0) ssum[tid >> 5] = tsum;
  __syncthreads();
  for (int s = 128; s > 0; s >>= 1) {
    if (tid < s) smax[tid] = fmaxf(smax[tid], smax[tid + s]);
    __syncthreads();
  }
  if (tid == 0) {
    float ts = 0.0f;
    for (int w = 0; w < 8; ++w) ts += ssum[w];
    scale[img] = (ts > 0.0f) ? (1.0f / (smax[0] + 1e-8f)) : 0.0f;
  }
}

// ---------------------------------------------------------------------------
// K4: fused sigmoid + four global sums (sumP, sumT, TP, sum(p*sdf)).
// Per-wave reduction via WMMA; per-block partials to workspace (fixed order,
// no float atomics -> deterministic).
// ---------------------------------------------------------------------------
__global__ void k_partials(const float* __restrict__ inp,
                           const int* __restrict__ tgt,
                           const float* __restrict__ d2,
                           const float* __restrict__ scale,
                           float* __restrict__ partials) {
  const int tid = threadIdx.x;
  const size_t stride = (size_t)1024 * 256;
  float sP = 0.0f, sT = 0.0f, sTP = 0.0f, sPS = 0.0f;
  for (size_t i = (size_t)blockIdx.x * 256 + tid; i < (size_t)NTOT_; i += stride) {
    const float p = 1.0f / (1.0f + expf(-inp[i]));
    const float t = (tgt[i] != 0) ? 1.0f : 0.0f;
    const float sdf =
        (sqrtf(d2[(size_t)NB_ * HW_ + i]) - sqrtf(d2[i])) * scale[i >> 18];
    sP  += p;
    sT  += t;
    sTP += p * t;
    sPS += p * sdf;
  }
  // uniform trip count (NTOT_ is a multiple of 1024*256) -> EXEC all ones here
  const float wP  = wave_sum32(sP);
  const float wT  = wave_sum32(sT);
  const float wTP = wave_sum32(sTP);
  const float wPS = wave_sum32(sPS);

  __shared__ float sred[8][4];
  if ((tid & 31) == 0) {
    const int wv = tid >> 5;
    sred[wv][0] = wP; sred[wv][1] = wT; sred[wv][2] = wTP; sred[wv][3] = wPS;
  }
  __syncthreads();
  if (tid == 0) {
    float a = 0, b = 0, c = 0, d = 0;
    for (int w = 0; w < 8; ++w) {
      a += sred[w][0]; b += sred[w][1]; c += sred[w][2]; d += sred[w][3];
    }
    const size_t o = (size_t)blockIdx.x * 4;
    partials[o + 0] = a; partials[o + 1] = b;
    partials[o + 2] = c; partials[o + 3] = d;
  }
}

// ---------------------------------------------------------------------------
// K5: sequential combine + tversky/focal/dice/boundary + NaN guards.
// ---------------------------------------------------------------------------
__global__ void k_final(const float* __restrict__ partials,
                        float* __restrict__ out) {
  if (threadIdx.x != 0 || blockIdx.x != 0) return;
  float sP = 0, sT = 0, sTP = 0, sPS = 0;
  for (int b = 0; b < 1024; ++b) {
    sP  += partials[b * 4 + 0];
    sT  += partials[b * 4 + 1];
    sTP += partials[b * 4 + 2];
    sPS += partials[b * 4 + 3];
  }
  const float TP = sTP, FP = sP - sTP, FN = sT - sTP;
  const float tversky = (TP + 1.0f) / (TP + 0.3f * FP + 0.7f * FN + 1.0f);
  float ft = powf(1.0f - tversky, 1.33f);
  const float dice = (2.0f * TP + 1.0f) / (sP + sT + 1.0f);
  float dl = 1.0f - dice;
  float bl = sPS / (float)NTOT_;
  ft = isnan(ft) ? 0.0f : ft;
  dl = isnan(dl) ? 0.0f : dl;
  bl = isnan(bl) ? 0.0f : bl;
  out[0] = 0.5f * ft + 0.3f * dl + 0.2f * bl;
}

// ---------------------------------------------------------------------------
// Launch. Workspace layout (floats):
//   [0,            8388608)  g2  : squared 1D row distances [mask][img][H][W]
//   [8388608,     16777216)  d2  : squared envelope results [mask][img][H][W]
//   [16777216,    16777232)  scale[16]
//   [16777232,    16781328)  block partials 1024*4
// Total ~67.1 MB.
// ---------------------------------------------------------------------------
extern "C" void kernel_launch(void* const* d_in, const int* in_sizes, int n_in,
                              void* d_out, int out_size, void* d_ws, size_t ws_size,
                              hipStream_t stream) {
  const float* inp = (const float*)d_in[0];
  const int*   tgt = (const int*)d_in[1];
  float* ws       = (float*)d_ws;
  float* g2       = ws;                       // 2 * 16 * HW
  float* d2       = ws + (size_t)8388608;
  float* scale    = ws + (size_t)16777216;
  float* partials = ws + (size_t)16777232;
  float* g2pos    = g2;
  float* g2neg    = g2 + (size_t)NB_ * HW_;

  k_rowdist <<<1024, 256, 0, stream>>>(tgt, g2pos, g2neg);
  k_envelope<<<dim3(16, 16, 32), 256, 0, stream>>>(g2, d2);
  k_imgstats<<<16, 256, 0, stream>>>(tgt, d2, scale);
  k_partials<<<1024, 256, 0, stream>>>(inp, tgt, d2, scale, partials);
  k_final   <<<1, 1, 0, stream>>>(partials, (float*)d_out);
}